// GCNNet_36455682409090
// MI455X (gfx1250) — compile-verified
//
#include <hip/hip_runtime.h>
#include <hip/hip_bf16.h>

typedef __attribute__((ext_vector_type(2))) float v2f;
typedef __attribute__((ext_vector_type(4))) float v4f;
typedef __attribute__((ext_vector_type(8))) float v8f;

// ---------------------------------------------------------------- utilities

__global__ void fill_f32(float* __restrict__ p, float val, long long n) {
    long long i = (long long)blockIdx.x * blockDim.x + threadIdx.x;
    if (i < n) p[i] = val;
}

__global__ void count_deg(const int* __restrict__ ei, float* __restrict__ deg, int E) {
    int e = blockIdx.x * blockDim.x + threadIdx.x;
    if (e < E) atomicAdd(&deg[ei[E + e]], 1.0f);   // row 1 of edge_index = dst
}

__global__ void rsqrt_inplace(float* __restrict__ deg, int n) {
    int i = blockIdx.x * blockDim.x + threadIdx.x;
    if (i < n) {
        float d = deg[i];
        deg[i] = d > 0.0f ? rsqrtf(d) : 0.0f;      // becomes dinv
    }
}

// ------------------------------------------------------- WMMA fp32 GEMM
// Out[M,NC] = A[M,K] @ W[K,NC].  Block = 256 threads = 8 waves.
// Each wave computes one 16x16 tile; block covers 32 rows x 64 cols.
// W is staged fully in LDS (K*NC*4 bytes <= 48KB here).
__global__ __launch_bounds__(256) void gemm_wmma_f32(
        const float* __restrict__ A, const float* __restrict__ W,
        float* __restrict__ Out, int M, int K, int NC)
{
    extern __shared__ float lds[];               // K * NC floats
    const int t = threadIdx.x;
    const int total = K * NC;                    // multiple of 1024 here
    for (int i = t * 4; i < total; i += 256 * 4)
        *(v4f*)&lds[i] = *(const v4f*)&W[i];
    __syncthreads();

    const int wave    = t >> 5;                  // 0..7
    const int lane    = t & 31;
    const int laneHi  = lane >> 4;               // 0 | 1 (K-half select)
    const int lane15  = lane & 15;
    const int row0    = blockIdx.x * 32 + (wave >> 2) * 16;
    const int col0    = blockIdx.y * 64 + (wave & 3) * 16;

    // A fragment row for this lane (clamped; M%32==0 in practice)
    int arow = row0 + lane15; if (arow >= M) arow = M - 1;
    const float* ap = A + (long long)arow * K;

    v8f acc = {};
    for (int k0 = 0; k0 < K; k0 += 4) {
        // A 16x4 f32 frag: lanes 0-15 -> K=k0,k0+1 ; lanes 16-31 -> K=k0+2,k0+3
        v2f a = *(const v2f*)&ap[k0 + laneHi * 2];
        // B 4x16 f32 frag (mirrored layout), read from LDS
        v2f b;
        b.x = lds[(k0 + laneHi * 2 + 0) * NC + col0 + lane15];
        b.y = lds[(k0 + laneHi * 2 + 1) * NC + col0 + lane15];
        acc = __builtin_amdgcn_wmma_f32_16x16x4_f32(
                false, a, false, b, (short)0, acc, false, false);
    }

    // D layout: VGPR i -> M = i (lanes 0-15) / M = 8+i (lanes 16-31), N = lane&15
    float* op = Out + (long long)row0 * NC + col0 + lane15;
#pragma unroll
    for (int i = 0; i < 8; ++i) {
        int m = i + laneHi * 8;
        if (row0 + m < M) op[(long long)m * NC] = acc[i];
    }
}

// ------------------------------------------------- edge gather/scatter
// One wave per edge (incl. self-loops e>=E). Features lane-strided.
__global__ __launch_bounds__(256) void aggregate(
        const float* __restrict__ H, const int* __restrict__ ei,
        const float* __restrict__ dinv, float* __restrict__ Out,
        int Nn, int E, int F)
{
    int gtid = blockIdx.x * 256 + threadIdx.x;
    int wave = gtid >> 5;
    int lane = gtid & 31;
    int total = E + Nn;
    if (wave >= total) return;
    int s, d;
    if (wave < E) { s = ei[wave]; d = ei[E + wave]; }
    else          { s = d = wave - E; }           // self-loop
    float norm = dinv[s] * dinv[d];
    const float* hs = H + (long long)s * F;
    float* od = Out + (long long)d * F;
    for (int f = lane; f < F; f += 32)
        atomicAdd(&od[f], hs[f] * norm);
}

__global__ void bias_relu(float* __restrict__ X, const float* __restrict__ b,
                          long long n, int F) {
    long long i = (long long)blockIdx.x * blockDim.x + threadIdx.x;
    if (i < n) {
        float v = X[i] + b[(int)(i % F)];
        X[i] = v > 0.0f ? v : 0.0f;
    }
}

// ------------------------------------------- fused bias + log_softmax (F=64)
// One wave per row; lane owns cols {lane, lane+32}; shfl_xor reductions.
__global__ __launch_bounds__(256) void bias_logsoftmax64(
        float* __restrict__ X, const float* __restrict__ b, int Nn)
{
    int gtid = blockIdx.x * 256 + threadIdx.x;
    int row = gtid >> 5;
    int lane = gtid & 31;
    if (row >= Nn) return;
    float* x = X + (long long)row * 64;
    float v0 = x[lane]      + b[lane];
    float v1 = x[lane + 32] + b[lane + 32];
    float m = fmaxf(v0, v1);
#pragma unroll
    for (int off = 16; off > 0; off >>= 1)
        m = fmaxf(m, __shfl_xor(m, off, 32));
    float s = expf(v0 - m) + expf(v1 - m);
#pragma unroll
    for (int off = 16; off > 0; off >>= 1)
        s += __shfl_xor(s, off, 32);
    float ls = logf(s);
    x[lane]      = v0 - m - ls;
    x[lane + 32] = v1 - m - ls;
}

// ---------------------------------------------------------------- launcher

extern "C" void kernel_launch(void* const* d_in, const int* in_sizes, int n_in,
                              void* d_out, int out_size, void* d_ws, size_t ws_size,
                              hipStream_t stream) {
    const float* x  = (const float*)d_in[0];
    const float* W1 = (const float*)d_in[1];
    const float* b1 = (const float*)d_in[2];
    const float* W2 = (const float*)d_in[3];
    const float* b2 = (const float*)d_in[4];
    const int*   ei = (const int*)d_in[5];

    const int F_IN = 96, H = 128, C = 64;
    const int Nn = in_sizes[0] / F_IN;           // 100000
    const int E  = in_sizes[5] / 2;              // 800000

    float* out_ls  = (float*)d_out;                       // [N, 64] logits->log_softmax
    float* out_emb = (float*)d_out + (long long)Nn * C;   // [N, 128] embedding

    float* deg  = (float*)d_ws;                  // N floats (becomes dinv)
    float* hbuf = deg + Nn;                      // N*128 floats, reused for layer 2

    // 1) degree with self-loops, then dinv = rsqrt(deg)
    fill_f32<<<(Nn + 255) / 256, 256, 0, stream>>>(deg, 1.0f, Nn);
    count_deg<<<(E + 255) / 256, 256, 0, stream>>>(ei, deg, E);
    rsqrt_inplace<<<(Nn + 255) / 256, 256, 0, stream>>>(deg, Nn);

    // 2) zero entire output (poisoned by harness); both regions receive atomics
    long long outN = (long long)Nn * (C + H);
    fill_f32<<<(int)((outN + 255) / 256), 256, 0, stream>>>((float*)d_out, 0.0f, outN);

    // 3) layer 1 GEMM: hbuf = x @ W1    (M=N, K=96, NC=128)
    {
        dim3 grid((Nn + 31) / 32, H / 64);
        gemm_wmma_f32<<<grid, 256, F_IN * H * sizeof(float), stream>>>(
            x, W1, hbuf, Nn, F_IN, H);
    }

    // 4) aggregate into embedding region, then bias+ReLU in place
    {
        long long waves = (long long)E + Nn;
        int blocks = (int)((waves * 32 + 255) / 256);
        aggregate<<<blocks, 256, 0, stream>>>(hbuf, ei, deg, out_emb, Nn, E, H);
        long long n = (long long)Nn * H;
        bias_relu<<<(int)((n + 255) / 256), 256, 0, stream>>>(out_emb, b1, n, H);
    }

    // 5) layer 2 GEMM: hbuf = emb @ W2  (M=N, K=128, NC=64)
    {
        dim3 grid((Nn + 31) / 32, C / 64);
        gemm_wmma_f32<<<grid, 256, H * C * sizeof(float), stream>>>(
            out_emb, W2, hbuf, Nn, H, C);
    }

    // 6) aggregate into logits region, then fused bias + log_softmax
    {
        long long waves = (long long)E + Nn;
        int blocks = (int)((waves * 32 + 255) / 256);
        aggregate<<<blocks, 256, 0, stream>>>(hbuf, ei, deg, out_ls, Nn, E, C);
        int rb = (Nn * 32 + 255) / 256;
        bias_logsoftmax64<<<rb, 256, 0, stream>>>(out_ls, b2, Nn);
    }
}